// SFI_MultiView_41008347742362
// MI455X (gfx1250) — compile-verified
//
#include <hip/hip_runtime.h>

typedef float v2f __attribute__((ext_vector_type(2)));
typedef float v8f __attribute__((ext_vector_type(8)));

#define DIM   256
#define NB    32
#define NC    5
#define NH    50
#define NL    32
#define NLEV  3
#define KTOP  10
#define THRESH 0.1f
#define EPSN  1e-12f

// ---------------------------------------------------------------------------
// Stage 1: proj = X @ W + b, then row-wise L2 normalize.
// Rows 0..159 come from cdd_repr, rows 160..1759 from his_repr (160 % 16 == 0,
// so each 16-row block is from exactly one source). One wave32 per block.
// GEMM via V_WMMA_F32_16X16X4_F32: 16 N-tiles x 64 K-steps.
// ---------------------------------------------------------------------------
__global__ __launch_bounds__(32) void proj_norm_kernel(
    const float* __restrict__ cdd, const float* __restrict__ his,
    const float* __restrict__ W,   const float* __restrict__ bias,
    float* __restrict__ c_norm, float* __restrict__ h_norm)
{
    __shared__ float tile[16 * DIM];   // 16 rows x 256 cols of projected data
    __shared__ float scl[16];

    const int lane = threadIdx.x;      // 0..31
    const int half = lane >> 4;        // 0 or 1
    const int m    = lane & 15;        // row/col within tile
    const int rb   = blockIdx.x * 16;  // global row base (0..1759)

    const float* src; float* dst; int srow;
    if (rb < 160) { src = cdd; dst = c_norm; srow = rb; }
    else          { src = his; dst = h_norm; srow = rb - 160; }

    const float* arow = src + (size_t)(srow + m) * DIM;

    for (int nt = 0; nt < 16; ++nt) {
        const int n0 = nt * 16;
        v8f acc = {};
        for (int k0 = 0; k0 < DIM; k0 += 4) {
            // A 16x4 layout: lanes 0-15 -> K=0,1 ; lanes 16-31 -> K=2,3
            const int ka = k0 + 2 * half;
            v2f a, b;
            a.x = arow[ka + 0];
            a.y = arow[ka + 1];
            // B 4x16 layout (mirrored): N = lane&15, same K split over halves
            b.x = W[(size_t)(ka + 0) * DIM + n0 + m];
            b.y = W[(size_t)(ka + 1) * DIM + n0 + m];
            acc = __builtin_amdgcn_wmma_f32_16x16x4_f32(
                      /*neg_a=*/false, a, /*neg_b=*/false, b,
                      /*c_mod=*/(short)0, acc,
                      /*reuse_a=*/false, /*reuse_b=*/false);
        }
        // C/D layout: VGPR i holds M=i (lanes 0-15) / M=i+8 (lanes 16-31), N=lane&15
        const float bn = bias[n0 + m];
#pragma unroll
        for (int i = 0; i < 8; ++i)
            tile[(i + 8 * half) * DIM + n0 + m] = acc[i] + bn;
    }
    __syncthreads();

    if (lane < 16) {
        const float* r = &tile[lane * DIM];
        float ss = 0.f;
        for (int j = 0; j < DIM; ++j) ss += r[j] * r[j];
        scl[lane] = 1.0f / fmaxf(sqrtf(ss), EPSN);
    }
    __syncthreads();

    // Coalesced vectorized writeback: 16*256 floats = 1024 float4
    float4*       d4 = (float4*)(dst + (size_t)srow * DIM);
    const float4* t4 = (const float4*)tile;
#pragma unroll 4
    for (int t = 0; t < 32; ++t) {
        const int v = t * 32 + lane;       // float4 index; 64 float4 per row
        const float s = scl[v >> 6];
        float4 x = t4[v];
        x.x *= s; x.y *= s; x.z *= s; x.w *= s;
        d4[v] = x;
    }
}

// ---------------------------------------------------------------------------
// Stage 2: attn[b,c,h] = c_norm[b,c,:] . h_norm[b,h,:], top-10 with threshold.
// One block per (b,c). Tiny.
// ---------------------------------------------------------------------------
__global__ __launch_bounds__(64) void attn_topk_kernel(
    const float* __restrict__ c_norm, const float* __restrict__ h_norm,
    float* __restrict__ w_out, int* __restrict__ idx_out)
{
    __shared__ float crow[DIM];
    __shared__ float attn[NH];

    const int bc = blockIdx.x;          // 0..159
    const int b  = bc / NC;
    const int t  = threadIdx.x;         // 0..63

    for (int j = t; j < DIM; j += 64) crow[j] = c_norm[(size_t)bc * DIM + j];
    __syncthreads();

    if (t < NH) {
        const float* hr = h_norm + (size_t)(b * NH + t) * DIM;
        float s = 0.f;
        for (int j = 0; j < DIM; ++j) s = fmaf(crow[j], hr[j], s);
        attn[t] = s;
    }
    __syncthreads();

    if (t == 0) {
        float vals[NH];
        for (int j = 0; j < NH; ++j) vals[j] = attn[j];
        for (int k = 0; k < KTOP; ++k) {
            float mx = -INFINITY; int mi = 0;
            for (int j = 0; j < NH; ++j)
                if (vals[j] > mx) { mx = vals[j]; mi = j; }  // first index on ties
            w_out[bc * KTOP + k]   = (mx < THRESH) ? 0.f : mx;
            idx_out[bc * KTOP + k] = mi;
            vals[mi] = -INFINITY;
        }
    }
}

// ---------------------------------------------------------------------------
// Stage 3: HBM-bound gather + scale. One block per (b,c,k): copies a
// 32*3*256 = 24576-float tile scaled by w, or zero-fills if masked.
// Pure coalesced float4 streaming: up to ~315 MB total -> ~13.5 us at 23.3 TB/s.
// ---------------------------------------------------------------------------
__global__ __launch_bounds__(256) void gather_scale_kernel(
    const float* __restrict__ emb, const float* __restrict__ w_in,
    const int* __restrict__ idx_in, float* __restrict__ out)
{
    const int blk  = blockIdx.x;                 // 0..1599 == bc*10 + k
    const int b    = blk / (NC * KTOP);
    const float w  = w_in[blk];
    const int hidx = idx_in[blk];

    const size_t inner = (size_t)NL * NLEV * DIM;         // 24576 floats
    const float4* s4 = (const float4*)(emb + ((size_t)b * NH + hidx) * inner);
    float4*       d4 = (float4*)(out + (size_t)blk * inner);
    const int tid = threadIdx.x;

    if (w == 0.f) {
        const float4 z = {0.f, 0.f, 0.f, 0.f};
#pragma unroll
        for (int i = 0; i < 24; ++i) d4[i * 256 + tid] = z;   // skip gather read
    } else {
#pragma unroll
        for (int i = 0; i < 24; ++i) {
            float4 x = s4[i * 256 + tid];
            x.x *= w; x.y *= w; x.z *= w; x.w *= w;
            d4[i * 256 + tid] = x;
        }
    }
}

// ---------------------------------------------------------------------------
extern "C" void kernel_launch(void* const* d_in, const int* in_sizes, int n_in,
                              void* d_out, int out_size, void* d_ws, size_t ws_size,
                              hipStream_t stream)
{
    const float* cdd  = (const float*)d_in[0];   // (32,5,256)
    const float* his  = (const float*)d_in[1];   // (32,50,256)
    const float* emb  = (const float*)d_in[2];   // (32,50,32,3,256)
    const float* W    = (const float*)d_in[3];   // (256,256)
    const float* bias = (const float*)d_in[4];   // (256,)

    float* ws     = (float*)d_ws;
    float* c_norm = ws;                               // 160*256 floats
    float* h_norm = ws + 160 * DIM;                   // 1600*256 floats
    float* w_ws   = h_norm + 1600 * DIM;              // 1600 floats
    int*   idx_ws = (int*)(w_ws + 1600);              // 1600 ints
    float* out    = (float*)d_out;                    // 39,321,600 floats

    proj_norm_kernel<<<110, 32, 0, stream>>>(cdd, his, W, bias, c_norm, h_norm);
    attn_topk_kernel<<<160, 64, 0, stream>>>(c_norm, h_norm, w_ws, idx_ws);
    gather_scale_kernel<<<1600, 256, 0, stream>>>(emb, w_ws, idx_ws, out);
}